// C3DLoss_89111981457692
// MI455X (gfx1250) — compile-verified
//
#include <hip/hip_runtime.h>
#include <hip/hip_bf16.h>
#include <math.h>

// ---------------------------------------------------------------------------
// C3D loss, fused single-pass tile kernel for gfx1250 (MI455X).
//   depth_pred [B,1,H,W] f32, depth_gt [B,1,H,W] f32, xy1_grid [B,3,H,W] f32,
//   K [B,3,3] f32 (unused), mask [B,1,H,W] bool (1 byte/elem).
// Output: single f32 scalar  -total / (n_valid + 1e-8).
// H=352, W=1216 are exact multiples of 32 -> every 32x32 tile interior is
// fully inside the image; only halos can go out of bounds (zero-filled, which
// exactly matches the reference's zero-padded shifts).
// ---------------------------------------------------------------------------

#define HH 352
#define WW 1216
#define TILE 32
#define PH 38   // TILE + 2*3 halo for xyz_p (normals on halo-2 need xyz at +-1)
#define NH 36   // TILE + 2*2 halo for n_p (window reaches +-2)
#define GH 34   // TILE + 2*1 halo for xyz_g (n_g needs xyz_g at +-1)

#define ELL_INV2 200.0f   // 1/(2*0.05^2)
#define RES_MIN 0.1f
#define RES_SPAN 1.9f     // RES_MAX - RES_MIN
#define EPSF 1e-8f

typedef __attribute__((ext_vector_type(2))) float v2f;
typedef __attribute__((ext_vector_type(8))) float v8f;

typedef __attribute__((address_space(1))) int as1_int;
typedef __attribute__((address_space(3))) int as3_int;

#if defined(__has_builtin)
#if __has_builtin(__builtin_amdgcn_global_load_async_to_lds_b32)
#define HAVE_ASYNC_LDS 1
#endif
#if __has_builtin(__builtin_amdgcn_s_wait_asynccnt)
#define HAVE_WAIT_ASYNC 1
#endif
#endif

__device__ __forceinline__ void wait_async_lds() {
#if defined(HAVE_ASYNC_LDS)
#if defined(HAVE_WAIT_ASYNC)
  __builtin_amdgcn_s_wait_asynccnt(0);
#else
  asm volatile("s_wait_asynccnt 0" ::: "memory");
#endif
#endif
}

// ---------------------------------------------------------------------------
// Deterministic block reduction of a (total, count) pair using
// V_WMMA_F32_16X16X4_F32 with a ones/selector B matrix.
//
// A (16x4 f32, 2 VGPR/lane):  lane L<16:  A[L][0]=a.x, A[L][1]=a.y
//                             lane 16+L:  A[L][2]=a.x, A[L][3]=a.y
// B (4x16, 2 VGPR/lane): VGPR0 = rows K=0 (lanes 0-15) / K=2 (lanes 16-31),
//                        VGPR1 = rows K=1 / K=3.
// Selector: B[even k][n] = (n<8),  B[odd k][n] = (n>=8)
//   => D[m][n<8]  = tot_m + tot_{m+16},  D[m][n>=8] = cnt_m + cnt_{m+16}
// After a per-lane sum of the 8 D regs and a second identical WMMA, the
// full-wave total lands in lane 0 (d[0]) and the full-wave count in lane 16.
// Must be called by all 256 threads with EXEC all-ones.
// ---------------------------------------------------------------------------
__device__ __forceinline__ float2 wmma_block_reduce2(float tot, float cnt,
                                                     float* s_red /*16 f32*/) {
  const int lane = threadIdx.x & 31;
  const int wid = threadIdx.x >> 5;

  v2f b;
  float sel = ((lane & 15) < 8) ? 1.0f : 0.0f;
  b.x = sel;           // rows K=0,2
  b.y = 1.0f - sel;    // rows K=1,3

  v2f a;
  a.x = tot;
  a.y = cnt;
  v8f c = {};
  v8f d = __builtin_amdgcn_wmma_f32_16x16x4_f32(false, a, false, b, (short)0,
                                                c, false, false);
  float s = ((d[0] + d[1]) + (d[2] + d[3])) + ((d[4] + d[5]) + (d[6] + d[7]));

  v2f a2;
  a2.x = s;
  a2.y = 0.0f;
  v8f d2 = __builtin_amdgcn_wmma_f32_16x16x4_f32(false, a2, false, b, (short)0,
                                                 c, false, false);
  // lanes 0-7 hold the wave total in every D reg; lanes 16-23 hold the count.
  if (lane == 0) s_red[wid] = d2[0];
  if (lane == 16) s_red[8 + wid] = d2[0];
  __syncthreads();

  float2 r;
  r.x = 0.0f;
  r.y = 0.0f;
  if (threadIdx.x == 0) {
    float T = 0.0f, C = 0.0f;
#pragma unroll
    for (int i = 0; i < 8; ++i) {
      T += s_red[i];
      C += s_red[8 + i];
    }
    r.x = T;
    r.y = C;
  }
  __syncthreads();  // allow s_red reuse by callers
  return r;
}

__device__ __forceinline__ void cross_norm(float ax, float ay, float az,
                                           float bx, float by, float bz,
                                           float& nx, float& ny, float& nz) {
  float cx = ay * bz - az * by;
  float cy = az * bx - ax * bz;
  float cz = ax * by - ay * bx;
  float inv = 1.0f / (sqrtf(cx * cx + cy * cy + cz * cz) + EPSF);
  nx = cx * inv;
  ny = cy * inv;
  nz = cz * inv;
}

__global__ __launch_bounds__(256) void c3d_tile_kernel(
    const float* __restrict__ dp, const float* __restrict__ dg,
    const float* __restrict__ xy1, const unsigned char* __restrict__ mask,
    float* __restrict__ ws, int ntx, int nty) {
  __shared__ float s_px[PH * PH], s_py[PH * PH], s_pz[PH * PH];
  __shared__ float s_nx[NH * NH], s_ny[NH * NH], s_nz[NH * NH];
  __shared__ float s_gx[GH * GH], s_gy[GH * GH], s_gz[GH * GH];
  __shared__ unsigned int s_mask[256];  // 32x32 bytes
  __shared__ float s_red[16];

  const int tid = threadIdx.x;
  const int blk = blockIdx.x;
  const int tilesPerB = ntx * nty;
  const int bb = blk / tilesPerB;
  const int t2 = blk - bb * tilesPerB;
  const int tyi = t2 / ntx;
  const int txi = t2 - tyi * ntx;
  const int gy0 = tyi * TILE;
  const int gx0 = txi * TILE;

  const size_t planeHW = (size_t)HH * WW;
  const float* dpb = dp + (size_t)bb * planeHW;
  const float* dgb = dg + (size_t)bb * planeHW;
  const float* x1b = xy1 + (size_t)bb * 3 * planeHW;
  const unsigned char* mb = mask + (size_t)bb * planeHW;

  // ---- async DMA of the 32x32 mask tile into LDS (interior always in-image)
  {
    int row = tid >> 3;        // 0..31
    int dc = (tid & 7) << 2;   // dword column in bytes
    const unsigned char* g = mb + (size_t)(gy0 + row) * WW + (gx0 + dc);
#if defined(HAVE_ASYNC_LDS)
    __builtin_amdgcn_global_load_async_to_lds_b32(
        (as1_int*)g, (as3_int*)&s_mask[tid], 0, 0);
#else
    s_mask[tid] = *(const unsigned int*)g;
#endif
  }

  // ---- stage 1: xyz_p (38x38, halo 3) and xyz_g (34x34, halo 1), zero OOB
  for (int i = tid; i < PH * PH; i += 256) {
    int ly = i / PH, lx = i - ly * PH;
    int gy = gy0 + ly - 3, gx = gx0 + lx - 3;
    float x = 0.0f, y = 0.0f, z = 0.0f;
    if (gy >= 0 && gy < HH && gx >= 0 && gx < WW) {
      size_t o = (size_t)gy * WW + gx;
      float d = dpb[o];
      x = x1b[o] * d;
      y = x1b[planeHW + o] * d;
      z = x1b[2 * planeHW + o] * d;
    }
    s_px[i] = x;
    s_py[i] = y;
    s_pz[i] = z;
  }
  for (int i = tid; i < GH * GH; i += 256) {
    int ly = i / GH, lx = i - ly * GH;
    int gy = gy0 + ly - 1, gx = gx0 + lx - 1;
    float x = 0.0f, y = 0.0f, z = 0.0f;
    if (gy >= 0 && gy < HH && gx >= 0 && gx < WW) {
      size_t o = (size_t)gy * WW + gx;
      float d = dgb[o];
      x = x1b[o] * d;
      y = x1b[planeHW + o] * d;
      z = x1b[2 * planeHW + o] * d;
    }
    s_gx[i] = x;
    s_gy[i] = y;
    s_gz[i] = z;
  }
  __syncthreads();

  // ---- stage 2: predicted normals on 36x36 (halo 2)
  for (int i = tid; i < NH * NH; i += 256) {
    int y2 = i / NH, x2 = i - y2 * NH;
    int p = (y2 + 1) * PH + (x2 + 1);
    float gxx = 0.5f * (s_px[p + 1] - s_px[p - 1]);
    float gxy = 0.5f * (s_py[p + 1] - s_py[p - 1]);
    float gxz = 0.5f * (s_pz[p + 1] - s_pz[p - 1]);
    float gyx = 0.5f * (s_px[p + PH] - s_px[p - PH]);
    float gyy = 0.5f * (s_py[p + PH] - s_py[p - PH]);
    float gyz = 0.5f * (s_pz[p + PH] - s_pz[p - PH]);
    float nx, ny, nz;
    cross_norm(gxx, gxy, gxz, gyx, gyy, gyz, nx, ny, nz);
    s_nx[i] = nx;
    s_ny[i] = ny;
    s_nz[i] = nz;
  }
  wait_async_lds();
  __syncthreads();

  // ---- stage 3: masked 5x5 correlation (skip ~95% of pixels)
  float tot = 0.0f, cnt = 0.0f;
  const unsigned char* mbytes = (const unsigned char*)s_mask;
  for (int i = tid; i < TILE * TILE; i += 256) {
    int ly = i >> 5, lx = i & 31;
    if (!mbytes[i]) continue;
    cnt += 1.0f;
    int gy = gy0 + ly, gx = gx0 + lx;

    int gc = (ly + 1) * GH + (lx + 1);
    float cx_ = s_gx[gc], cy_ = s_gy[gc], cz_ = s_gz[gc];
    // gt normal on the fly (zero-padded central differences)
    float gxx = 0.5f * (s_gx[gc + 1] - s_gx[gc - 1]);
    float gxy = 0.5f * (s_gy[gc + 1] - s_gy[gc - 1]);
    float gxz = 0.5f * (s_gz[gc + 1] - s_gz[gc - 1]);
    float gyx = 0.5f * (s_gx[gc + GH] - s_gx[gc - GH]);
    float gyy = 0.5f * (s_gy[gc + GH] - s_gy[gc - GH]);
    float gyz = 0.5f * (s_gz[gc + GH] - s_gz[gc - GH]);
    float ngx, ngy, ngz;
    cross_norm(gxx, gxy, gxz, gyx, gyy, gyz, ngx, ngy, ngz);

    float acc = 0.0f;
#pragma unroll
    for (int dy = -2; dy <= 2; ++dy) {
      int iy = gy + dy;
      bool vy = (iy >= 0) && (iy < HH);
      int pp = (ly + 3 + dy) * PH + (lx + 3);
      int nn = (ly + 2 + dy) * NH + (lx + 2);
#pragma unroll
      for (int dx = -2; dx <= 2; ++dx) {
        int ix = gx + dx;
        if (!vy || ix < 0 || ix >= WW) continue;  // vs == 0
        int pi = pp + dx, ni = nn + dx;
        float ex = s_px[pi] - cx_;
        float ey = s_py[pi] - cy_;
        float ez = s_pz[pi] - cz_;
        float d2 = ex * ex + ey * ey + ez * ez;
        float kg = __expf(-d2 * ELL_INV2);
        float nk = fabsf(s_nx[ni] * ngx + s_ny[ni] * ngy + s_nz[ni] * ngz);
        acc += kg * (RES_MIN + RES_SPAN * nk);
      }
    }
    tot += acc;
  }

  // ---- stage 4: deterministic WMMA block reduce, partial per block to ws
  float2 r = wmma_block_reduce2(tot, cnt, s_red);
  if (tid == 0) {
    ws[2 * blk] = r.x;
    ws[2 * blk + 1] = r.y;
  }
}

__global__ __launch_bounds__(256) void c3d_final_kernel(
    const float* __restrict__ ws, int nblk, float* __restrict__ out) {
  __shared__ float s_red[16];
  float tot = 0.0f, cnt = 0.0f;
  for (int i = threadIdx.x; i < nblk; i += 256) {
    tot += ws[2 * i];
    cnt += ws[2 * i + 1];
  }
  float2 r = wmma_block_reduce2(tot, cnt, s_red);
  if (threadIdx.x == 0) out[0] = -r.x / (r.y + EPSF);
}

extern "C" void kernel_launch(void* const* d_in, const int* in_sizes, int n_in,
                              void* d_out, int out_size, void* d_ws,
                              size_t ws_size, hipStream_t stream) {
  (void)n_in;
  (void)out_size;
  (void)ws_size;
  const float* dp = (const float*)d_in[0];
  const float* dg = (const float*)d_in[1];
  const float* xy1 = (const float*)d_in[2];
  // d_in[3] = K (unused by the dense path)
  const unsigned char* mask = (const unsigned char*)d_in[4];
  float* out = (float*)d_out;
  float* ws = (float*)d_ws;

  const int Bn = in_sizes[0] / (HH * WW);
  const int ntx = WW / TILE;  // 38
  const int nty = HH / TILE;  // 11
  const int nblk = Bn * ntx * nty;

  c3d_tile_kernel<<<nblk, 256, 0, stream>>>(dp, dg, xy1, mask, ws, ntx, nty);
  c3d_final_kernel<<<1, 256, 0, stream>>>(ws, nblk, out);
}